// LearnedEWMA_18700287607233
// MI455X (gfx1250) — compile-verified
//
#include <hip/hip_runtime.h>
#include <hip/hip_bf16.h>
#include <math.h>

// ---------------------------------------------------------------------------
// LearnedEWMA on MI455X (gfx1250): reformulate the EWMA scan as a tall-skinny
// f32 GEMM  E = X(4096x8192) * V(8192x8)  and run it on the WMMA pipe.
// ---------------------------------------------------------------------------

typedef __attribute__((ext_vector_type(2))) float v2f;
typedef __attribute__((ext_vector_type(8))) float v8f;

#define B_ROWS 4096
#define T_LEN  8192
#define N_WIN  8
#define TILE_M 16
#define WAVES  8
#define BLOCK  256
#define KSTEPS_TOTAL    (T_LEN / 4)            // 2048 WMMA K-steps of 4
#define KSTEPS_PER_WAVE (KSTEPS_TOTAL / WAVES) // 256 per wave

// ---------------------------------------------------------------------------
// Kernel A: precompute weight matrix V[t,w] directly in WMMA B-fragment
// lane order.  Workspace layout: idx = s*64 + lane*2 + j, where step s covers
// times 4s..4s+3, and (lane,j) maps to the assumed f32 4x16 B layout:
//   VGPR j, lane L:  K_local = 2*(L/16) + j,  N = L%16.
// V[0,w] = d^(T-1); V[t,w] = (1-d)*d^(T-1-t); zero for padded windows w>=8.
// ---------------------------------------------------------------------------
__global__ void ewma_weights_kernel(const float* __restrict__ log_halflife,
                                    float* __restrict__ wgtB) {
  int idx = blockIdx.x * blockDim.x + threadIdx.x;   // 0 .. 2048*64-1
  if (idx >= KSTEPS_TOTAL * 64) return;
  int s    = idx >> 6;
  int r    = idx & 63;
  int lane = r >> 1;
  int j    = r & 1;
  int t    = 4 * s + 2 * (lane >> 4) + j;
  int n    = lane & 15;
  float val = 0.0f;
  if (n < N_WIN) {
    float hl = expf(log_halflife[n]);
    hl = fminf(fmaxf(hl, 1.0f), 200.0f);
    float invhl = 1.0f / hl;
    float d = exp2f(-invhl);
    float age = (float)(T_LEN - 1 - t);
    val = exp2f(-age * invhl);          // d^(T-1-t)
    if (t > 0) val *= (1.0f - d);
  }
  wgtB[idx] = val;
}

// ---------------------------------------------------------------------------
// Kernel B: one workgroup per 16-row tile of X.  8 waves split the K=8192
// dimension; each wave runs 256 x V_WMMA_F32_16X16X4_F32 accumulating a
// 16x16 f32 fragment, partials are reduced through LDS, then the signal
// (ewma - last)/(|last|+1e-8) is applied.
// ---------------------------------------------------------------------------
__global__ void __launch_bounds__(BLOCK)
ewma_wmma_kernel(const float* __restrict__ X,
                 const float* __restrict__ wgtB,
                 float* __restrict__ out) {
  __shared__ float red[WAVES * 256];

  const int tile = blockIdx.x;            // 0..255 (4096/16)
  const int lane = threadIdx.x & 31;
  const int wv   = threadIdx.x >> 5;

  // A-fragment addressing per ISA 16x4 f32 A layout:
  //   lane L: M = L%16, K_local = 2*(L/16) + {0,1} -> aligned b64 load.
  const int row    = tile * TILE_M + (lane & 15);
  const int kStart = wv * KSTEPS_PER_WAVE * 4;    // this wave's time offset
  const float* ap = X + (size_t)row * T_LEN + kStart + 2 * (lane >> 4);
  const float* bp = wgtB + (size_t)(wv * KSTEPS_PER_WAVE) * 64 + lane * 2;

  v8f acc = {};
#pragma unroll 8
  for (int s = 0; s < KSTEPS_PER_WAVE; ++s) {
    v2f a = *(const v2f*)ap;   // 8B aligned: kStart%1024==0, offset even
    v2f b = *(const v2f*)bp;   // L2-resident weight fragments
    // 8 args: (neg_a, A, neg_b, B, c_mod, C, reuse_a, reuse_b)
    acc = __builtin_amdgcn_wmma_f32_16x16x4_f32(
        false, a, false, b, (short)0, acc, false, false);
    ap += 4;
    bp += 64;
  }

  // Reduce the 8 per-wave 16x16 fragments through LDS.
#pragma unroll
  for (int r = 0; r < 8; ++r)
    red[wv * 256 + lane * 8 + r] = acc[r];
  __syncthreads();

  const int i = threadIdx.x;   // 0..255 indexes (srcLane, reg) pairs
  float sum = 0.0f;
#pragma unroll
  for (int w = 0; w < WAVES; ++w) sum += red[w * 256 + i];

  // C/D layout: VGPR r, lane L -> M = r + 8*(L/16), N = L%16.
  const int srcLane = i >> 3;
  const int r = i & 7;
  const int M = r + 8 * (srcLane >> 4);
  const int N = srcLane & 15;
  if (N < N_WIN) {
    const int grow = tile * TILE_M + M;
    const float last = X[(size_t)grow * T_LEN + (T_LEN - 1)];
    out[grow * N_WIN + N] = (sum - last) / (fabsf(last) + 1e-8f);
  }
}

// ---------------------------------------------------------------------------
// Launch
// ---------------------------------------------------------------------------
extern "C" void kernel_launch(void* const* d_in, const int* in_sizes, int n_in,
                              void* d_out, int out_size, void* d_ws, size_t ws_size,
                              hipStream_t stream) {
  const float* price        = (const float*)d_in[0];  // (4096, 8192) f32
  const float* log_halflife = (const float*)d_in[1];  // (8,) f32
  float* out  = (float*)d_out;                        // (4096, 8) f32
  float* wgtB = (float*)d_ws;                         // 2048*64 f32 = 512 KB

  (void)in_sizes; (void)n_in; (void)out_size; (void)ws_size;

  const int wElems = KSTEPS_TOTAL * 64;               // 131072
  ewma_weights_kernel<<<(wElems + BLOCK - 1) / BLOCK, BLOCK, 0, stream>>>(
      log_halflife, wgtB);

  ewma_wmma_kernel<<<B_ROWS / TILE_M, BLOCK, 0, stream>>>(price, wgtB, out);
}